// AffinityBiFC_51118700757141
// MI455X (gfx1250) — compile-verified
//
#include <hip/hip_runtime.h>
#include <hip/hip_bf16.h>

// Problem sizes (fixed by the reference)
#define BB 4
#define NN 128
#define MM 128
#define DD 256
#define BD 1024

typedef __attribute__((ext_vector_type(2))) float v2f;
typedef __attribute__((ext_vector_type(4))) float v4f;
typedef __attribute__((ext_vector_type(8))) float v8f;

// ---------------------------------------------------------------------------
// Kernel 1: Aw[d*D+e] = sum_k A[d,e,k] * W[k]
// A is 268 MB -> this kernel is the bandwidth roofline (~11.5 us @ 23.3 TB/s).
// One wave32 per (d,e) pair; each wave streams 4 KB contiguous (8 float4/lane).
// A is single-pass and larger than the 192 MB L2, so load it NON-TEMPORAL to
// avoid evicting the small hot working set (X/Y/Aw/T) between kernels.
// W (4 KB) staged in LDS once per block so HBM traffic is A only.
// ---------------------------------------------------------------------------
__global__ void __launch_bounds__(256)
fold_w_kernel(const float* __restrict__ A, const float* __restrict__ W,
              float* __restrict__ Aw) {
  __shared__ v4f w4[BD / 4];
  const int tid = threadIdx.x;          // 0..255
  w4[tid] = ((const v4f*)W)[tid];       // 256 * 16B = 4 KB
  __syncthreads();

  const int wave = tid >> 5;            // 0..7
  const int lane = tid & 31;
  const size_t pair = (size_t)blockIdx.x * 8 + wave;   // 0..65535 (= d*D+e)
  const v4f* a4 = (const v4f*)A + pair * (BD / 4);

  float acc = 0.0f;
#pragma unroll
  for (int j = 0; j < 8; ++j) {
    const v4f a = __builtin_nontemporal_load(&a4[lane + 32 * j]);  // TH=NT stream
    const v4f w = w4[lane + 32 * j];
    acc = fmaf(a.x, w.x, acc);
    acc = fmaf(a.y, w.y, acc);
    acc = fmaf(a.z, w.z, acc);
    acc = fmaf(a.w, w.w, acc);
  }
  // wave32 tree reduction
#pragma unroll
  for (int off = 16; off >= 1; off >>= 1)
    acc += __shfl_xor(acc, off, 32);
  if (lane == 0) Aw[pair] = acc;        // tiny, reused next kernel: keep RT
}

// ---------------------------------------------------------------------------
// Kernel 2: T[r, e] = X[r, :] @ Aw[:, e]   (r = flattened b*N+n, 512x256x256)
// One wave per 16x16 output tile; V_WMMA_F32_16X16X4_F32, K-loop step 4.
// ---------------------------------------------------------------------------
__global__ void __launch_bounds__(32)
gemm_xaw_kernel(const float* __restrict__ X, const float* __restrict__ Aw,
                float* __restrict__ T) {
  const int lane = threadIdx.x & 31;
  const int half = lane >> 4;           // 0: lanes 0-15, 1: lanes 16-31
  const int l16  = lane & 15;
  const int row0 = blockIdx.x * 16;     // over 512 rows (B*N)
  const int col0 = blockIdx.y * 16;     // over 256 cols (e)

  v8f c = {};
  for (int k = 0; k < DD; k += 4) {
    // A frag (16x4): lane holds X[row0+l16][k + 2*half + {0,1}]
    const float* xp = X + (size_t)(row0 + l16) * DD + k + 2 * half;
    v2f a; a.x = xp[0]; a.y = xp[1];
    // B frag (4x16): vgpr r holds Aw[(k + r + 2*half)][col0 + l16]
    const float* bp = Aw + (size_t)(k + 2 * half) * DD + col0 + l16;
    v2f bf; bf.x = bp[0]; bf.y = bp[DD];
    c = __builtin_amdgcn_wmma_f32_16x16x4_f32(false, a, false, bf,
                                              (short)0, c, false, false);
  }
  // D: vgpr r -> row (row0 + r + 8*half), col (col0 + l16)
#pragma unroll
  for (int r = 0; r < 8; ++r)
    T[(size_t)(row0 + r + 8 * half) * DD + col0 + l16] = c[r];
}

// ---------------------------------------------------------------------------
// Kernel 3: S[b,n,m] = T[b,n,:] . Y[b,m,:] + bias    (per-batch 128x256x128^T)
// B-matrix is Y accessed "transposed": B[K=e][N=m] = Y[b][m][e], which is a
// contiguous float2 per lane since the 2 VGPR elements differ only in e.
// ---------------------------------------------------------------------------
__global__ void __launch_bounds__(32)
gemm_ty_kernel(const float* __restrict__ T, const float* __restrict__ Y,
               const float* __restrict__ bias, float* __restrict__ S) {
  const int lane = threadIdx.x & 31;
  const int half = lane >> 4;
  const int l16  = lane & 15;
  const int b  = blockIdx.z;
  const int n0 = blockIdx.x * 16;
  const int m0 = blockIdx.y * 16;
  const float* Tb = T + (size_t)b * NN * DD;
  const float* Yb = Y + (size_t)b * MM * DD;

  v8f c = {};
  for (int k = 0; k < DD; k += 4) {
    const float* ap = Tb + (size_t)(n0 + l16) * DD + k + 2 * half;
    v2f a; a.x = ap[0]; a.y = ap[1];
    const float* bp = Yb + (size_t)(m0 + l16) * DD + k + 2 * half;
    v2f bf; bf.x = bp[0]; bf.y = bp[1];
    c = __builtin_amdgcn_wmma_f32_16x16x4_f32(false, a, false, bf,
                                              (short)0, c, false, false);
  }
  const float bv = bias[0];
#pragma unroll
  for (int r = 0; r < 8; ++r)
    S[((size_t)b * NN + n0 + r + 8 * half) * MM + m0 + l16] = c[r] + bv;
}

// ---------------------------------------------------------------------------
extern "C" void kernel_launch(void* const* d_in, const int* in_sizes, int n_in,
                              void* d_out, int out_size, void* d_ws, size_t ws_size,
                              hipStream_t stream) {
  (void)in_sizes; (void)n_in; (void)out_size; (void)ws_size;
  const float* X = (const float*)d_in[0];   // [B, N, D]
  const float* Y = (const float*)d_in[1];   // [B, M, D]
  const float* A = (const float*)d_in[2];   // [D, D, BD]
  const float* W = (const float*)d_in[3];   // [1, BD]
  const float* b = (const float*)d_in[4];   // [1]
  float* out = (float*)d_out;               // [B, N, M]

  float* Aw = (float*)d_ws;                 // D*D floats        = 256 KB
  float* T  = Aw + (size_t)DD * DD;         // B*N*D floats      = 512 KB

  // 1) Fold W into A: 65536 (d,e) pairs, 8 per block.
  fold_w_kernel<<<(DD * DD) / 8, 256, 0, stream>>>(A, W, Aw);
  // 2) T = X @ Aw : 32 x 16 tiles of 16x16, one wave each.
  gemm_xaw_kernel<<<dim3((BB * NN) / 16, DD / 16), 32, 0, stream>>>(X, Aw, T);
  // 3) S = T @ Y^T + b : per batch 8 x 8 tiles.
  gemm_ty_kernel<<<dim3(NN / 16, MM / 16, BB), 32, 0, stream>>>(T, Y, b, out);
}